// CustomKernel_1In1Out_Conv2D_33895881900750
// MI455X (gfx1250) — compile-verified
//
#include <hip/hip_runtime.h>
#include <cstdint>

// ---------------------------------------------------------------------------
// Conv2D 16x16 VALID, fp32, via V_WMMA_F32_16X16X4_F32 on gfx1250 (MI455X).
//
// O[b, y, x] = sum_{i,j} X[b, y+i, x+j] * K[i,j]   (cross-correlation, as XLA)
//
// Per 16x16 output tile: one K=512 matmul done as 128 chained wmma 16x16x4:
//   A[m, k]   = patch[m+i][t]            (t = 4c + khalf + {0,1})
//   B[k, n]   = Kpad[i][t - n + 15]      (Toeplitz of kernel row i, zero-padded)
//   D[m, n]  += A x B                    (fp32 accumulate, exact-precision path)
//
// Memory-bound problem (~0.53 GB @ 23.3 TB/s ~= 23 us); this design reads each
// input element once into LDS per tile (31x31 halo reuse) and keeps all math
// in the WMMA pipe. Patch staging uses global_load_async_to_lds_b128 +
// s_wait_asynccnt (ASYNCcnt path, CDNA5 ISA ch.10/15.18).
//
// Assumed wave32 operand layouts (CDNA5 ISA 7.12.2):
//   A 16x4 f32 : lane l<16 -> M=l, K={0,1} in v0,v1 ; lane l>=16 -> M=l-16, K={2,3}
//   B 4x16 f32 : lane l<16 -> N=l, K={0,1} in v0,v1 ; lane l>=16 -> N=l-16, K={2,3}
//   C/D 16x16  : VGPR v, lane l: M = v + 8*(l>=16), N = l%16
// ---------------------------------------------------------------------------

#define BATCH 64
#define H     1024
#define W     1024
#define KH    16
#define KW    16
#define OH    1009
#define OW    1009
#define TILE  16
#define TY_N  64          // ceil(1009/16)
#define TX_N  64
#define WAVES_PER_BLOCK 8
#define KP_W  48          // padded Toeplitz row length (needs 47, round to 48)

typedef float v2f __attribute__((ext_vector_type(2)));
typedef float v8f __attribute__((ext_vector_type(8)));
typedef int   v4i __attribute__((vector_size(16)));   // matches builtin param type

__global__ __launch_bounds__(256) void conv16x16_wmma_f32(
    const float* __restrict__ x, const float* __restrict__ kern,
    float* __restrict__ out)
{
  __shared__ float kp[KH * KP_W];                    // 3 KB  (shared by block)
  __shared__ float patch[WAVES_PER_BLOCK][32 * 32];  // 4 KB per wave (32 KB)

  const int tid  = threadIdx.x;
  const int lane = tid & 31;
  const int wv   = tid >> 5;

  // ---- Toeplitz source rows: kp[i][p] = K[i][p-15] for 15<=p<=30, else 0
  for (int idx = tid; idx < KH * KP_W; idx += blockDim.x) {
    const int i = idx / KP_W, p = idx % KP_W;
    kp[idx] = (p >= 15 && p <= 30) ? kern[i * KW + (p - 15)] : 0.0f;
  }

  // ---- one wave == one 16x16 output tile
  const int tile = blockIdx.x * WAVES_PER_BLOCK + wv;
  const int b    = tile >> 12;          // / (64*64)
  const int rem  = tile & 4095;
  const int y0   = (rem >> 6) * TILE;
  const int x0   = (rem & 63) * TILE;

  // ---- stage 32x32 input patch: rows y0..y0+31 (clamped), cols x0..x0+31
  // (clamped to col<=1020 in 16B chunks; clamped slots provably only feed
  //  Toeplitz zeros or invalid outputs, since x0 % 16 == 0)
  const int    row    = min(y0 + lane, H - 1);
  const size_t inBase = (size_t)b * (size_t)(H * W) + (size_t)row * W;
  float*       prow   = &patch[wv][lane * 32];

#if defined(__gfx1250__) && __has_builtin(__builtin_amdgcn_global_load_async_to_lds_b128)
  #pragma unroll
  for (int c4 = 0; c4 < 8; ++c4) {
    const int col = min(x0 + c4 * 4, W - 4);
    const float* gp = x + inBase + col;
    float*       lp = prow + c4 * 4;
    __builtin_amdgcn_global_load_async_to_lds_b128(
        (__attribute__((address_space(1))) v4i*)(uintptr_t)gp,
        (__attribute__((address_space(3))) v4i*)(uint32_t)(uintptr_t)lp,
        0, 0);
  }
  #if __has_builtin(__builtin_amdgcn_s_wait_asynccnt)
  __builtin_amdgcn_s_wait_asynccnt(0);
  #else
  asm volatile("s_wait_asynccnt 0" ::: "memory");
  #endif
#else
  #pragma unroll
  for (int c4 = 0; c4 < 8; ++c4) {
    const int col = min(x0 + c4 * 4, W - 4);
    *(float4*)(prow + c4 * 4) = *(const float4*)(x + inBase + col);
  }
#endif

  __syncthreads();   // kp visibility (and patch, in fallback path)

  // ---- 128 chained wmma_f32_16x16x4_f32, all operands from LDS
  v8f acc = {0.f, 0.f, 0.f, 0.f, 0.f, 0.f, 0.f, 0.f};

  const int khalf2 = (lane >> 4) << 1;                 // 0 (K=0,1) or 2 (K=2,3)
  const int aBase  = (lane & 15) * 32 + khalf2;        // float index in patch
  const int bBase  = (15 - (lane & 15)) + khalf2;      // float index in kp row
  const float* pw  = &patch[wv][0];

  for (int i = 0; i < KH; ++i) {
    #pragma unroll
    for (int c = 0; c < 8; ++c) {
      // A: patch[(m+i)][4c + khalf + {0,1}] -> consecutive -> ds b64
      v2f a = *(const v2f*)(pw + aBase + i * 32 + c * 4);
      // B: kp[i][4c + khalf + {0,1} + 15 - n]
      const int bi = bBase + i * KP_W + c * 4;
      v2f bv; bv.x = kp[bi]; bv.y = kp[bi + 1];
      acc = __builtin_amdgcn_wmma_f32_16x16x4_f32(
          false, a, false, bv, (short)0, acc, false, false);
    }
  }

  // ---- store valid outputs (C/D layout -> (y0+v+8*hi, x0+lane%16))
  const size_t ob = (size_t)b * (size_t)OH * (size_t)OW;
  const int    n  = lane & 15;
  const int    mh = (lane >> 4) * 8;
  #pragma unroll
  for (int v = 0; v < 8; ++v) {
    const int yy = y0 + v + mh;
    const int xx = x0 + n;
    if (yy < OH && xx < OW) out[ob + (size_t)yy * OW + xx] = acc[v];
  }
}

extern "C" void kernel_launch(void* const* d_in, const int* in_sizes, int n_in,
                              void* d_out, int out_size, void* d_ws, size_t ws_size,
                              hipStream_t stream) {
  (void)in_sizes; (void)n_in; (void)d_ws; (void)ws_size; (void)out_size;
  const float* x = (const float*)d_in[0];
  const float* k = (const float*)d_in[1];
  float*       o = (float*)d_out;
  const int total_tiles = BATCH * TY_N * TX_N;           // 262144
  const int blocks      = total_tiles / WAVES_PER_BLOCK; // 32768
  conv16x16_wmma_f32<<<blocks, 256, 0, stream>>>(x, k, o);
}